// Attention_75471165325727
// MI455X (gfx1250) — compile-verified
//
#include <hip/hip_runtime.h>

typedef __attribute__((ext_vector_type(2))) float v2f;
typedef __attribute__((ext_vector_type(8))) float v8f;
typedef __attribute__((ext_vector_type(4))) unsigned uint4v;
typedef __attribute__((ext_vector_type(8))) unsigned uint8v;

#define HDIM 256
#define SDIM 4096
#define BDIM 32
#define KSEL 64
#define EPSV 1e-7f
#define TPW  4   // row-tiles per wave (amortizes W_p LDS staging)

// ---------------------------------------------------------------------------
// Kernel 1: ap[b,s] = sigmoid( tanh(enc[b,s,:] @ W_p) @ v_p )
// W_p staged into LDS by one Tensor-Data-Mover DMA (tensor_load_to_lds),
// GEMM via V_WMMA_F32_16X16X4_F32 (full fp32 precision), 4 tiles per wave.
// ---------------------------------------------------------------------------
__global__ __launch_bounds__(256) void k_proj(const float* __restrict__ enc,
                                              const float* __restrict__ Wp,
                                              const float* __restrict__ vp,
                                              float* __restrict__ ap) {
    extern __shared__ float smem[];          // [HDIM*HDIM] W_p, then [HDIM] v_p
    float* wp_s = smem;                      // dynamic LDS starts at offset 0
    float* vp_s = smem + HDIM * HDIM;

    const int tid = threadIdx.x;
    const int wave = tid >> 5;
    const int lane = tid & 31;
    const int half = lane >> 4;   // 0: K pair {0,1} / rows 0..7 ; 1: K pair {2,3} / rows 8..15
    const int l16  = lane & 15;

    // --- TDM: DMA the whole 256x256 f32 W_p tensor into LDS (one descriptor) ---
    if (wave == 0) {
        const unsigned long long ga = (unsigned long long)Wp;
        // D# group 0: count=1 | lds_addr=0 | global_addr | type=2("image")
        uint4v g0;
        g0.x = 0x1u;                                   // count=1, user mode
        g0.y = 0u;                                     // lds_addr = 0 (bytes)
        g0.z = (unsigned)(ga & 0xFFFFFFFFu);           // global_addr[31:0]
        g0.w = (unsigned)(((ga >> 32) & 0x01FFFFFFu) | 0x80000000u); // addr[56:32] | type=2
        // D# group 1: data_size=4B, tensor 256x256, tile 256x256, stride0=256
        uint8v g1;
        g1.s0 = 0x2u << 16;        // workgroup_mask=0, data_size=2 (4 bytes)
        g1.s1 = 256u << 16;        // tensor_dim0[15:0] in [31:16]
        g1.s2 = 256u << 16;        // tensor_dim0[31:16]=0, tensor_dim1[15:0]=256
        g1.s3 = 256u << 16;        // tensor_dim1[31:16]=0, tile_dim0=256
        g1.s4 = 256u;              // tile_dim1=256, tile_dim2=0
        g1.s5 = 256u;              // tensor_dim0_stride[31:0] = 256 elements
        g1.s6 = 0u;                // stride0[47:32]=0, dim1_stride[15:0]=0
        g1.s7 = 1u;                // dim1_stride[47:16] = 1 -> stride1 = 65536
        asm volatile("tensor_load_to_lds %0, %1" :: "s"(g0), "s"(g1) : "memory");
    }
    // v_p is tiny: normal cooperative load
    for (int i = tid; i < HDIM; i += 256) vp_s[i] = vp[i];
    if (wave == 0) __builtin_amdgcn_s_wait_tensorcnt(0);
    __syncthreads();

    const long tile0 = ((long)blockIdx.x * 8 + wave) * TPW;

    for (int t = 0; t < TPW; ++t) {
        const long r0 = (tile0 + t) * 16;

        v8f acc[16] = {};   // 16 N-tiles of 16x16 f32 accumulators

        // A-fragment stream: lane holds row (r0+l16), K cols (4*kt + 2*half, +1)
        const float* arow = enc + (r0 + l16) * HDIM + half * 2;
        const float2* arow2 = (const float2*)arow;

        for (int kt = 0; kt < 64; ++kt) {
            if ((kt & 15) == 0)
                __builtin_prefetch(arow + (kt + 16) * 4, 0, 1);  // global_prefetch_b8

            float2 a2 = arow2[kt * 2];           // enc[row][4*kt + 2*half .. +1]
            v2f A; A.x = a2.x; A.y = a2.y;

            const int kk = kt * 4 + half * 2;    // K rows this lane-half supplies
#pragma unroll
            for (int nt = 0; nt < 16; ++nt) {
                v2f Bf;
                Bf.x = wp_s[(kk + 0) * HDIM + nt * 16 + l16];
                Bf.y = wp_s[(kk + 1) * HDIM + nt * 16 + l16];
                acc[nt] = __builtin_amdgcn_wmma_f32_16x16x4_f32(
                    false, A, false, Bf, (short)0, acc[nt], false, false);
            }
        }

        // Epilogue: row-sum of tanh(y)*v_p over H, then sigmoid.
        // C/D layout: lane half h, VGPR r -> element (M = r + 8h, N = nt*16 + l16)
        float part[8];
#pragma unroll
        for (int r = 0; r < 8; ++r) {
            float s = 0.f;
#pragma unroll
            for (int nt = 0; nt < 16; ++nt)
                s += tanhf(acc[nt][r]) * vp_s[nt * 16 + l16];
            part[r] = s;
        }
#pragma unroll
        for (int r = 0; r < 8; ++r) {
            float s = part[r];
            s += __shfl_xor(s, 1, 32);   // xor masks < 16 stay within each 16-lane half
            s += __shfl_xor(s, 2, 32);
            s += __shfl_xor(s, 4, 32);
            s += __shfl_xor(s, 8, 32);
            if (l16 == 0) {
                const long row = r0 + r + half * 8;
                ap[row] = 1.f / (1.f + expf(-s));
            }
        }
    }
}

// ---------------------------------------------------------------------------
// Kernel 2: per batch, exact top-64 threshold via 4-pass radix select on the
// float bit pattern (all values positive => uint-ordered). Writes the row
// scale factor (1 + a/(a+eps) for selected rows, else 1) and weights == 1.
// ---------------------------------------------------------------------------
__global__ __launch_bounds__(256) void k_topk(const float* __restrict__ ap,
                                              float* __restrict__ factor,
                                              float* __restrict__ weights) {
    __shared__ float vals[SDIM];
    __shared__ int hist[256];
    __shared__ int sel_bin;
    __shared__ int sel_rem;

    const int b = blockIdx.x;
    const int tid = threadIdx.x;
    const float* apb = ap + (long)b * SDIM;

    for (int i = tid; i < SDIM; i += 256) vals[i] = apb[i];
    __syncthreads();

    unsigned prefix = 0u;
    int remaining = KSEL;

    for (int pass = 0; pass < 4; ++pass) {
        const int shift = 24 - pass * 8;
        hist[tid] = 0;
        __syncthreads();

        for (int i = tid; i < SDIM; i += 256) {
            unsigned u = __float_as_uint(vals[i]);
            bool match = (pass == 0) || ((u >> (shift + 8)) == (prefix >> (shift + 8)));
            if (match) atomicAdd(&hist[(u >> shift) & 255], 1);
        }
        __syncthreads();

        if (tid == 0) {
            int cum = 0, bin = 0;
            for (int bb = 255; bb >= 0; --bb) {
                cum += hist[bb];
                if (cum >= remaining) { bin = bb; cum -= hist[bb]; break; }
            }
            sel_bin = bin;             // byte of the 64th-largest at this position
            sel_rem = remaining - cum; // rank within the surviving group
        }
        __syncthreads();
        prefix |= ((unsigned)sel_bin) << shift;
        remaining = sel_rem;
        __syncthreads();
    }

    const float T = __uint_as_float(prefix);  // exact value of the 64th largest
    const int nEqualAllowed = remaining;      // #elements == T that are in top-k

    for (int i = tid; i < SDIM; i += 256) {
        float v = vals[i];
        bool in;
        if (v > T) {
            in = true;
        } else if (v == T) {
            int eqBefore = 0;                 // jax top_k tie-break: lower index first
            for (int j = 0; j < i; ++j) eqBefore += (vals[j] == T);
            in = (eqBefore < nEqualAllowed);
        } else {
            in = false;
        }
        const float f = in ? (1.f + v / (v + EPSV)) : 1.f;
        factor[(long)b * SDIM + i] = f;
        weights[(long)b * SDIM + i] = 1.f;    // softmax over singleton axis == 1
    }
}

// ---------------------------------------------------------------------------
// Kernel 3: context = enc * factor[row]  (float4 streaming; enc likely L2-hot)
// ---------------------------------------------------------------------------
__global__ __launch_bounds__(256) void k_scale(const float* __restrict__ enc,
                                               const float* __restrict__ factor,
                                               float* __restrict__ outc) {
    const long i4 = (long)blockIdx.x * blockDim.x + threadIdx.x;
    const long n4 = (long)BDIM * SDIM * (HDIM / 4);
    if (i4 >= n4) return;
    const long row = i4 / (HDIM / 4);
    const float f = factor[row];
    const float4 v = ((const float4*)enc)[i4];
    float4 o;
    o.x = v.x * f; o.y = v.y * f; o.z = v.z * f; o.w = v.w * f;
    ((float4*)outc)[i4] = o;
}

// ---------------------------------------------------------------------------
extern "C" void kernel_launch(void* const* d_in, const int* in_sizes, int n_in,
                              void* d_out, int out_size, void* d_ws, size_t ws_size,
                              hipStream_t stream) {
    (void)in_sizes; (void)n_in; (void)out_size; (void)ws_size;

    const float* enc = (const float*)d_in[0];   // (32, 4096, 256)
    // d_in[1] = decoder_out, d_in[4] = W_a: dead inputs (softmax over singleton axis)
    const float* Wp  = (const float*)d_in[2];   // (256, 256)
    const float* vp  = (const float*)d_in[3];   // (256, 1)

    float* outc = (float*)d_out;                            // context: B*S*H
    float* outw = (float*)d_out + (long)BDIM * SDIM * HDIM; // weights: B*S

    float* ap     = (float*)d_ws;               // B*S floats
    float* factor = ap + (long)BDIM * SDIM;     // B*S floats

    const size_t lds_bytes = (size_t)(HDIM * HDIM + HDIM) * sizeof(float); // 263,168 B

    const int tiles  = (BDIM * SDIM) / 16;      // 8192 row-tiles
    const int blocks = tiles / (8 * TPW);       // 8 waves/block, TPW tiles/wave

    k_proj<<<dim3(blocks), dim3(256), lds_bytes, stream>>>(enc, Wp, vp, ap);
    k_topk<<<dim3(BDIM), dim3(256), 0, stream>>>(ap, factor, outw);

    const long n4 = (long)BDIM * SDIM * (HDIM / 4);
    k_scale<<<dim3((unsigned)((n4 + 255) / 256)), dim3(256), 0, stream>>>(enc, factor, outc);
}